// GAT_26938034881119
// MI455X (gfx1250) — compile-verified
//
#include <hip/hip_runtime.h>
#include <hip/hip_bf16.h>
#include <math.h>

// ---------------- problem constants (from the reference) ----------------
#define NODES 50000
#define IN_DIM 128
#define NEDGE 800000
#define ETOT (NEDGE + NODES)       // reference appends self loops
#define NEG_SLOPE 0.2f
#define BN_EPS 1e-5f
#define PADROWS (((NODES + 127) / 128) * 128)   // 50048: GEMM outputs padded -> unguarded stores

typedef float v2f __attribute__((ext_vector_type(2)));
typedef float v8f __attribute__((ext_vector_type(8)));

// ---------------- small utility kernels ----------------
__global__ void fill_f32(float* p, float v, int n) {
  int i = blockIdx.x * blockDim.x + threadIdx.x;
  if (i < n) p[i] = v;
}

__global__ void init_bias(float* out, const float* __restrict__ bias, int n, int C) {
  int i = blockIdx.x * blockDim.x + threadIdx.x;
  if (i < n) out[i] = bias[i % C];
}

// ---------------- f32 WMMA GEMM:  C[PADROWS x NC] = A[nrows x 128] * W[128 x NC] ----------------
// 256 threads = 8 waves; each wave owns a 16-row strip; W + a 128-row A tile staged in LDS.
// Cout must have >= gridDim.x*128 rows (workspace is padded) -> stores are unconditional.
template <int NC>
__global__ void gemm_wmma(const float* __restrict__ A, const float* __restrict__ W,
                          float* __restrict__ Cout, int nrows) {
  extern __shared__ float smem[];
  constexpr int WS = NC + 16;   // W stride: lane-half 1 (rows k0+2,k0+3) lands 32 banks away
  constexpr int XS = 130;       // x stride: keeps 8B alignment, spreads rows over banks by 2
  float* Wl = smem;             // [128][WS]
  float* xs = smem + 128 * WS;  // [128][XS]

  const int tid  = threadIdx.x;
  const int row0 = blockIdx.x * 128;

  // stage W (128 x NC), coalesced
  for (int idx = tid; idx < 128 * NC; idx += 256) {
    int k = idx / NC, n = idx % NC;
    Wl[k * WS + n] = W[idx];
  }
  // stage A tile (128 x 128), coalesced, zero-pad past nrows
  for (int idx = tid; idx < 128 * 128; idx += 256) {
    int r = idx >> 7, c = idx & 127;
    int g = row0 + r;
    xs[r * XS + c] = (g < nrows) ? A[(size_t)g * 128 + c] : 0.0f;
  }
  __syncthreads();

  const int wave  = tid >> 5;
  const int lane  = tid & 31;
  const int lhalf = lane >> 4;   // 0: lanes 0-15 (K=k0,k0+1), 1: lanes 16-31 (K=k0+2,k0+3)
  const int l16   = lane & 15;
  const int mrow  = wave * 16 + l16;

  for (int nt = 0; nt < NC / 16; ++nt) {
    v8f acc = {};
#pragma unroll 4
    for (int k0 = 0; k0 < 128; k0 += 4) {
      int kb = k0 + 2 * lhalf;
      // A fragment: 16x4, per-lane {A[m][kb], A[m][kb+1]}  (8B aligned LDS read)
      v2f a = *(const v2f*)&xs[mrow * XS + kb];
      // B fragment: 4x16, per-lane {W[kb][n], W[kb+1][n]}
      v2f b;
      b.x = Wl[(kb + 0) * WS + nt * 16 + l16];
      b.y = Wl[(kb + 1) * WS + nt * 16 + l16];
      acc = __builtin_amdgcn_wmma_f32_16x16x4_f32(false, a, false, b, (short)0, acc,
                                                  false, false);
    }
    // C layout: VGPR r -> M=r (lanes 0-15), M=r+8 (lanes 16-31); N = lane&15
#pragma unroll
    for (int r = 0; r < 8; ++r) {
      int gr = row0 + wave * 16 + r + 8 * lhalf;
      Cout[(size_t)gr * NC + nt * 16 + l16] = acc[r];   // unguarded: padded buffer
    }
  }
}

// ---------------- per-(node,head) attention coefficients ----------------
// a_src[n,h] = sum_c h[n,h,c]*att_src[h,c] ; same for a_dst. 64 threads / block.
template <int HD>
__global__ void attn_coef(const float* __restrict__ h, const float* __restrict__ att_s,
                          const float* __restrict__ att_d, float* __restrict__ as_,
                          float* __restrict__ ad_) {
  __shared__ float ss[64], sd[64];
  int node = blockIdx.x / HD, hd = blockIdx.x % HD;
  int t = threadIdx.x;
  float v = h[(size_t)node * HD * 64 + hd * 64 + t];
  ss[t] = v * att_s[hd * 64 + t];
  sd[t] = v * att_d[hd * 64 + t];
  __syncthreads();
  for (int s = 32; s > 0; s >>= 1) {
    if (t < s) { ss[t] += ss[t + s]; sd[t] += sd[t + s]; }
    __syncthreads();
  }
  if (t == 0) {
    as_[node * HD + hd] = ss[0];
    ad_[node * HD + hd] = sd[0];
  }
}

// ---------------- edge-phase helpers ----------------
__device__ __forceinline__ void edge_nodes(const int* __restrict__ ei, int i, int& s, int& d) {
  if (i < NEDGE) { s = ei[i]; d = ei[NEDGE + i]; }   // edge_index[0]=src, [1]=dst
  else           { s = d = i - NEDGE; }              // appended self loops
}

__device__ __forceinline__ float leaky(float e) { return e > 0.0f ? e : NEG_SLOPE * e; }

// float atomic max via sign-split int/uint atomics (valid with -inf init)
__device__ __forceinline__ void atomicMaxF(float* addr, float val) {
  if (val >= 0.0f) atomicMax((int*)addr, __float_as_int(val));
  else             atomicMin((unsigned int*)addr, (unsigned int)__float_as_int(val));
}

template <int HD>
__global__ void edge_max(const int* __restrict__ ei, const float* __restrict__ as_,
                         const float* __restrict__ ad_, float* m) {
  int t = blockIdx.x * blockDim.x + threadIdx.x;
  if (t >= ETOT * HD) return;
  int i = t / HD, hd = t % HD;
  int s, d; edge_nodes(ei, i, s, d);
  float e = leaky(as_[s * HD + hd] + ad_[d * HD + hd]);
  atomicMaxF(&m[d * HD + hd], e);
}

template <int HD>
__global__ void edge_sum(const int* __restrict__ ei, const float* __restrict__ as_,
                         const float* __restrict__ ad_, const float* __restrict__ m,
                         float* den) {
  int t = blockIdx.x * blockDim.x + threadIdx.x;
  if (t >= ETOT * HD) return;
  int i = t / HD, hd = t % HD;
  int s, d; edge_nodes(ei, i, s, d);
  float e = leaky(as_[s * HD + hd] + ad_[d * HD + hd]);
  atomicAdd(&den[d * HD + hd], expf(e - m[d * HD + hd]));
}

// materialize alpha[e,h] once so the scatter does no redundant transcendental work
template <int HD>
__global__ void edge_alpha(const int* __restrict__ ei, const float* __restrict__ as_,
                           const float* __restrict__ ad_, const float* __restrict__ m,
                           const float* __restrict__ den, float* __restrict__ alpha) {
  int t = blockIdx.x * blockDim.x + threadIdx.x;
  if (t >= ETOT * HD) return;
  int i = t / HD, hd = t % HD;
  int s, d; edge_nodes(ei, i, s, d);
  float e = leaky(as_[s * HD + hd] + ad_[d * HD + hd]);
  alpha[t] = expf(e - m[d * HD + hd]) / (den[d * HD + hd] + 1e-16f);
}

// one block per edge, HD*64 threads = one lane per channel; +bias folded via out pre-init
template <int HD>
__global__ void edge_scatter(const int* __restrict__ ei, const float* __restrict__ h,
                             const float* __restrict__ alpha, float* out) {
  int i = blockIdx.x;
  int t = threadIdx.x;           // HD*64 threads
  int hd = t >> 6, c = t & 63;
  int s, d; edge_nodes(ei, i, s, d);
  float a = alpha[i * HD + hd];  // broadcast within the 64-lane group
  atomicAdd(&out[(size_t)d * HD * 64 + hd * 64 + c],
            h[(size_t)s * HD * 64 + hd * 64 + c] * a);
}

// ---------------- BatchNorm (batch stats) + ELU ----------------
__global__ void bn_accum(const float* __restrict__ x, float* bnsum, float* bnsq) {
  int c  = threadIdx.x;            // 128 channels, coalesced across the block
  int r0 = blockIdx.x * 256;
  int r1 = min(r0 + 256, NODES);
  float s = 0.f, q = 0.f;
  for (int r = r0; r < r1; ++r) {
    if (r + 32 < r1) __builtin_prefetch(&x[(size_t)(r + 32) * 128 + c], 0, 1);
    float v = x[(size_t)r * 128 + c];
    s += v; q += v * v;
  }
  atomicAdd(&bnsum[c], s);
  atomicAdd(&bnsq[c], q);
}

__global__ void bn_apply_elu(float* x, const float* __restrict__ bnsum,
                             const float* __restrict__ bnsq,
                             const float* __restrict__ gamma,
                             const float* __restrict__ beta) {
  int i = blockIdx.x * blockDim.x + threadIdx.x;
  if (i >= NODES * 128) return;
  int c = i & 127;
  float mu  = bnsum[c] * (1.0f / NODES);
  float var = bnsq[c] * (1.0f / NODES) - mu * mu;   // biased var, like jnp.var
  float y = (x[i] - mu) * rsqrtf(var + BN_EPS) * gamma[c] + beta[c];
  x[i] = y > 0.f ? y : expm1f(y);                    // ELU(alpha=1)
}

// ---------------- host orchestration ----------------
extern "C" void kernel_launch(void* const* d_in, const int* in_sizes, int n_in,
                              void* d_out, int out_size, void* d_ws, size_t ws_size,
                              hipStream_t stream) {
  const float* x      = (const float*)d_in[0];
  const int*   ei     = (const int*)d_in[1];   // jax default: int32 (x64 disabled)
  const float* W0     = (const float*)d_in[2];
  const float* attS0  = (const float*)d_in[3];
  const float* attD0  = (const float*)d_in[4];
  const float* bias0  = (const float*)d_in[5];
  const float* gamma0 = (const float*)d_in[6];
  const float* beta0  = (const float*)d_in[7];
  const float* W1     = (const float*)d_in[8];
  const float* attS1  = (const float*)d_in[9];
  const float* attD1  = (const float*)d_in[10];
  const float* bias1  = (const float*)d_in[11];
  float* out = (float*)d_out;

  char* wp = (char*)d_ws;
  auto carve = [&](size_t bytes) -> float* {
    float* p = (float*)wp;
    wp += (bytes + 255) & ~(size_t)255;
    return p;
  };
  float* h0     = carve((size_t)PADROWS * 128 * 4); // layer0 linear output (padded rows)
  float* agg0   = carve((size_t)PADROWS * 128 * 4); // layer0 aggregated; BN/ELU in place
  float* h1     = carve((size_t)PADROWS * 64 * 4);  // layer1 linear output (padded rows)
  float* as0    = carve((size_t)NODES * 2 * 4);
  float* ad0    = carve((size_t)NODES * 2 * 4);
  float* m0     = carve((size_t)NODES * 2 * 4);
  float* dn0    = carve((size_t)NODES * 2 * 4);
  float* alpha0 = carve((size_t)ETOT * 2 * 4);
  float* as1    = carve((size_t)NODES * 4);
  float* ad1    = carve((size_t)NODES * 4);
  float* m1     = carve((size_t)NODES * 4);
  float* dn1    = carve((size_t)NODES * 4);
  float* alpha1 = carve((size_t)ETOT * 4);
  float* bnsum  = carve(128 * 4);
  float* bnsq   = carve(128 * 4);

  const int gemmBlocks = PADROWS / 128;                           // 391
  const size_t sh0 = (size_t)(128 * (128 + 16) + 128 * 130) * 4;  // ~140 KB LDS
  const size_t sh1 = (size_t)(128 * (64 + 16) + 128 * 130) * 4;   // ~105 KB LDS

  // ---- layer 0 (2 heads x 64, concat) ----
  gemm_wmma<128><<<gemmBlocks, 256, sh0, stream>>>(x, W0, h0, NODES);
  attn_coef<2><<<NODES * 2, 64, 0, stream>>>(h0, attS0, attD0, as0, ad0);
  fill_f32<<<(NODES * 2 + 255) / 256, 256, 0, stream>>>(m0, -INFINITY, NODES * 2);
  fill_f32<<<(NODES * 2 + 255) / 256, 256, 0, stream>>>(dn0, 0.f, NODES * 2);
  init_bias<<<(NODES * 128 + 255) / 256, 256, 0, stream>>>(agg0, bias0, NODES * 128, 128);
  edge_max<2><<<(ETOT * 2 + 255) / 256, 256, 0, stream>>>(ei, as0, ad0, m0);
  edge_sum<2><<<(ETOT * 2 + 255) / 256, 256, 0, stream>>>(ei, as0, ad0, m0, dn0);
  edge_alpha<2><<<(ETOT * 2 + 255) / 256, 256, 0, stream>>>(ei, as0, ad0, m0, dn0, alpha0);
  edge_scatter<2><<<ETOT, 128, 0, stream>>>(ei, h0, alpha0, agg0);

  // ---- BatchNorm + ELU (in place on agg0) ----
  fill_f32<<<1, 256, 0, stream>>>(bnsum, 0.f, 128);
  fill_f32<<<1, 256, 0, stream>>>(bnsq, 0.f, 128);
  bn_accum<<<(NODES + 255) / 256, 128, 0, stream>>>(agg0, bnsum, bnsq);
  bn_apply_elu<<<(NODES * 128 + 255) / 256, 256, 0, stream>>>(agg0, bnsum, bnsq, gamma0, beta0);

  // ---- layer 1 (1 head x 64, mean == identity), aggregate straight into d_out ----
  gemm_wmma<64><<<gemmBlocks, 256, sh1, stream>>>(agg0, W1, h1, NODES);
  attn_coef<1><<<NODES, 64, 0, stream>>>(h1, attS1, attD1, as1, ad1);
  fill_f32<<<(NODES + 255) / 256, 256, 0, stream>>>(m1, -INFINITY, NODES);
  fill_f32<<<(NODES + 255) / 256, 256, 0, stream>>>(dn1, 0.f, NODES);
  init_bias<<<(NODES * 64 + 255) / 256, 256, 0, stream>>>(out, bias1, NODES * 64, 64);
  edge_max<1><<<(ETOT + 255) / 256, 256, 0, stream>>>(ei, as1, ad1, m1);
  edge_sum<1><<<(ETOT + 255) / 256, 256, 0, stream>>>(ei, as1, ad1, m1, dn1);
  edge_alpha<1><<<(ETOT + 255) / 256, 256, 0, stream>>>(ei, as1, ad1, m1, dn1, alpha1);
  edge_scatter<1><<<ETOT, 64, 0, stream>>>(ei, h1, alpha1, out);
}